// HungerModulatedPolicy_36163624633171
// MI455X (gfx1250) — compile-verified
//
#include <hip/hip_runtime.h>
#include <hip/hip_bf16.h>

typedef __attribute__((ext_vector_type(2))) float v2f;
typedef __attribute__((ext_vector_type(8))) float v8f;

#define IN_DIM   512
#define HIDDEN   100000
#define OUT_DIM  128
#define NNZ      6400000

// ---------------------------------------------------------------------------
// Init: zero agg, seed y with b_out (d_out/d_ws are poisoned before timing and
// we accumulate atomically, so this must run every launch).
// ---------------------------------------------------------------------------
__global__ __launch_bounds__(256)
void init_kernel(const float* __restrict__ b_out,
                 float* __restrict__ agg, float* __restrict__ y) {
    int i = blockIdx.x * blockDim.x + threadIdx.x;
    if (i < HIDDEN)  agg[i] = 0.0f;
    if (i < OUT_DIM) y[i]   = b_out[i];
}

// ---------------------------------------------------------------------------
// Stage 1: h = relu(W_in @ x + b_in), via V_WMMA_F32_16X16X4_F32.
// One wave handles 16 rows. A = 16x4 f32 tile of W_in (2 VGPRs/lane: lanes
// 0-15 hold K=0,1 of row M=lane; lanes 16-31 hold K=2,3). B broadcasts
// x[k..k+3] to all 16 columns, so every column of D is the same dot product.
// ---------------------------------------------------------------------------
__global__ __launch_bounds__(256)
void gemv_in_kernel(const float* __restrict__ W, const float* __restrict__ x,
                    const float* __restrict__ b, float* __restrict__ h) {
    const int lane = threadIdx.x & 31;
    const int wave = blockIdx.x * (blockDim.x >> 5) + (threadIdx.x >> 5);
    const int row0 = wave * 16;
    if (row0 >= HIDDEN) return;          // wave-uniform: EXEC all-1 for WMMA

    const int mr   = lane & 15;          // row within 16-row tile
    const int koff = (lane >> 4) << 1;   // 0 for lanes 0-15, 2 for lanes 16-31
    const float* arow = W + (size_t)(row0 + mr) * IN_DIM + koff;

    v8f c = {};
    #pragma unroll 4
    for (int k = 0; k < IN_DIM; k += 4) {
        v2f a  = *(const v2f*)(arow + k);        // W[row][k+koff], W[row][k+koff+1]
        v2f bb = *(const v2f*)(x + k + koff);    // broadcast x chunk across columns
        c = __builtin_amdgcn_wmma_f32_16x16x4_f32(
                false, a, false, bb, (short)0, c, false, false);
    }

    // D layout: lanes 0-15 hold M=0..7 in VGPRs 0..7; lanes 16-31 hold M=8..15.
    // All columns equal -> lane 0 and lane 16 carry the 16 row results.
    if (mr == 0) {
        const int base = row0 + ((lane >> 4) << 3);
        #pragma unroll
        for (int i = 0; i < 8; ++i) {
            float v = c[i] + b[base + i];
            h[base + i] = v > 0.0f ? v : 0.0f;
        }
    }
}

// ---------------------------------------------------------------------------
// Stage 2: COO scatter  agg[rows[e]] += edge_w[e] * h[cols[e]].
// h/agg are 400 KB each -> resident in L2; hardware f32 atomics via
// unsafeAtomicAdd (global_atomic_add_f32).
// ---------------------------------------------------------------------------
__global__ __launch_bounds__(256)
void edge_kernel(const int* __restrict__ rows, const int* __restrict__ cols,
                 const float* __restrict__ w, const float* __restrict__ h,
                 float* __restrict__ agg) {
    int e = blockIdx.x * blockDim.x + threadIdx.x;
    if (e < NNZ) {
        float m = w[e] * h[cols[e]];
        unsafeAtomicAdd(&agg[rows[e]], m);
    }
}

// ---------------------------------------------------------------------------
// Stage 3: y += W_out @ relu(agg). 8 row-tiles x 40 K-chunks of 2500
// (100000 = 40*2500 exactly); partial tiles accumulated with f32 atomics
// onto y (already seeded with b_out).
// ---------------------------------------------------------------------------
__global__ __launch_bounds__(256)
void gemv_out_kernel(const float* __restrict__ W, const float* __restrict__ agg,
                     float* __restrict__ y) {
    const int lane    = threadIdx.x & 31;
    const int wave    = blockIdx.x * (blockDim.x >> 5) + (threadIdx.x >> 5);
    const int rowTile = wave & 7;        // 0..7  -> rows rowTile*16 .. +15
    const int chunk   = wave >> 3;       // 0..39 -> K range chunk*2500 .. +2499
    const int row0    = rowTile * 16;
    const int k0      = chunk * 2500;

    const int mr   = lane & 15;
    const int koff = (lane >> 4) << 1;
    const float* arow = W + (size_t)(row0 + mr) * HIDDEN + koff;

    v8f c = {};
    #pragma unroll 4
    for (int k = k0; k < k0 + 2500; k += 4) {
        v2f a = *(const v2f*)(arow + k);
        float g0 = agg[k + koff];
        float g1 = agg[k + koff + 1];
        v2f bb;
        bb.x = g0 > 0.0f ? g0 : 0.0f;    // relu fused into B operand
        bb.y = g1 > 0.0f ? g1 : 0.0f;
        c = __builtin_amdgcn_wmma_f32_16x16x4_f32(
                false, a, false, bb, (short)0, c, false, false);
    }

    if (mr == 0) {
        const int base = row0 + ((lane >> 4) << 3);
        #pragma unroll
        for (int i = 0; i < 8; ++i)
            unsafeAtomicAdd(&y[base + i], c[i]);
    }
}

// ---------------------------------------------------------------------------
// Inputs (setup_inputs order): x, W_in, b_in, rows, cols, edge_w, W_out, b_out
// ---------------------------------------------------------------------------
extern "C" void kernel_launch(void* const* d_in, const int* in_sizes, int n_in,
                              void* d_out, int out_size, void* d_ws, size_t ws_size,
                              hipStream_t stream) {
    const float* x      = (const float*)d_in[0];
    const float* W_in   = (const float*)d_in[1];
    const float* b_in   = (const float*)d_in[2];
    const int*   rows   = (const int*)  d_in[3];
    const int*   cols   = (const int*)  d_in[4];
    const float* edge_w = (const float*)d_in[5];
    const float* W_out  = (const float*)d_in[6];
    const float* b_out  = (const float*)d_in[7];
    float*       y      = (float*)d_out;

    float* h   = (float*)d_ws;           // HIDDEN floats
    float* agg = h + 102400;             // HIDDEN floats (padded offset)

    init_kernel<<<(HIDDEN + 255) / 256, 256, 0, stream>>>(b_out, agg, y);

    // 100000/16 = 6250 waves; 8 waves per 256-thread block -> 782 blocks.
    gemv_in_kernel<<<(6250 + 7) / 8, 256, 0, stream>>>(W_in, x, b_in, h);

    edge_kernel<<<NNZ / 256, 256, 0, stream>>>(rows, cols, edge_w, h, agg);

    // 8 row-tiles * 40 chunks = 320 waves -> 40 blocks of 256 threads.
    gemv_out_kernel<<<40, 256, 0, stream>>>(W_out, agg, y);
}